// ScaledDotProductAttention_5720896438775
// MI455X (gfx1250) — compile-verified
//
#include <hip/hip_runtime.h>

// ScaledDotProductAttention for B=16, S=2048, D=64 (f32), returning (output, attn).
// attn (B*S*S f32 = 256MB) is a required output, so scores are materialized once:
// kept in LDS (128KB of the 320KB WGP LDS) for an exact single-pass softmax.
// Matrix math uses V_WMMA_F32_16X16X4_F32 to match the f32 reference precision.

typedef float v2f __attribute__((ext_vector_type(2)));
typedef float v4f __attribute__((ext_vector_type(4)));
typedef float v8f __attribute__((ext_vector_type(8)));

#define S_LEN   2048
#define D_DIM   64
#define NEG_INF (-1e30f)

__global__ void __launch_bounds__(256)
sdpa_f32_wmma_kernel(const float* __restrict__ q,
                     const float* __restrict__ k,
                     const float* __restrict__ v,
                     const unsigned char* __restrict__ mask,  // jax bool: 1 byte/elem
                     float* __restrict__ out,                 // [B,S,D]
                     float* __restrict__ attn)                // [B,S,S]
{
    extern __shared__ float smem[];
    float* s_scores = smem;                    // 16 x 2048 f32 = 128KB
    float* s_q      = smem + 16 * S_LEN;       // 16 x 64
    float* s_out    = s_q  + 16 * D_DIM;       // 16 x 64 accumulator

    const int wg = blockIdx.x;                 // B * S/16 = 2048 blocks
    const int b  = wg >> 7;                    // / (S/16)
    const int q0 = (wg & 127) << 4;            // query row block start

    const int tid  = threadIdx.x;              // 0..255 (8 wave32)
    const int wave = tid >> 5;
    const int lane = tid & 31;

    const float* qb = q + ((size_t)b * S_LEN + q0) * D_DIM;
    const float* kb = k + (size_t)b * S_LEN * D_DIM;
    const float* vb = v + (size_t)b * S_LEN * D_DIM;

    // ---- stage Q tile (16x64) into LDS; zero the output accumulator ----
    for (int i = tid; i < 16 * D_DIM; i += 256) {
        s_q[i]   = qb[i];
        s_out[i] = 0.0f;
    }
    __syncthreads();

    // ---- per-lane A fragments of Q for V_WMMA_F32_16X16X4_F32 ----
    // A 16x4 layout: lanes 0-15 hold row M=lane {K=4s, 4s+1}; lanes 16-31 hold
    // row M=lane-16 {K=4s+2, 4s+3}.  float2 loads are 8B-aligned.
    const int mrow = lane & 15;
    const int c0   = (lane < 16) ? 0 : 2;
    v2f aq[16];
#pragma unroll
    for (int s = 0; s < 16; ++s)
        aq[s] = *(const v2f*)&s_q[mrow * D_DIM + 4 * s + c0];

    // =========== Phase 1: scores = (Q K^T)/8 -> LDS row ===========
    // 128 K-tiles of 16 columns; each wave owns 16 tiles, processed in pairs
    // (two independent WMMA accumulate chains sharing the A fragments).
    const int mofs = (lane < 16) ? 0 : 8;
    for (int t = 0; t < 16; t += 2) {
        const int n0a = (wave * 16 + t) << 4;
        const int n0b = n0a + 16;
        v8f acca = {};
        v8f accb = {};
#pragma unroll
        for (int s = 0; s < 16; ++s) {
            // B 4x16 layout mirrors A: row n=lane&15 of K, cols 4s+c0(+1)
            v2f bka = *(const v2f*)&kb[(size_t)(n0a + mrow) * D_DIM + 4 * s + c0];
            v2f bkb = *(const v2f*)&kb[(size_t)(n0b + mrow) * D_DIM + 4 * s + c0];
            acca = __builtin_amdgcn_wmma_f32_16x16x4_f32(
                false, aq[s], false, bka, (short)0, acca, false, false);
            accb = __builtin_amdgcn_wmma_f32_16x16x4_f32(
                false, aq[s], false, bkb, (short)0, accb, false, false);
        }
        // C/D layout: acc[i] holds (m = i + mofs, n = lane&15)
#pragma unroll
        for (int i = 0; i < 8; ++i) {
            s_scores[(i + mofs) * S_LEN + n0a + mrow] = acca[i] * 0.125f;
            s_scores[(i + mofs) * S_LEN + n0b + mrow] = accb[i] * 0.125f;
        }
    }
    __syncthreads();

    // =========== Phase 2: mask + row softmax in LDS + stream attn ===========
    // 16 lanes per row (row = tid/16); each lane owns 4 consecutive columns
    // per step -> b128 LDS ops, coalesced uint mask loads, b128 attn stores.
    {
        const int r = tid >> 4;
        const int l = tid & 15;
        float* srow = s_scores + r * S_LEN;
        const unsigned char* mrowp = mask + ((size_t)b * S_LEN + q0 + r) * S_LEN;

        float mx = -1e38f;
        for (int j = 0; j < S_LEN / 64; ++j) {
            const int c = 4 * l + 64 * j;
            v4f sv = *(v4f*)&srow[c];
            const unsigned int mw = *(const unsigned int*)&mrowp[c];  // 4 bools
            if (mw & 0x000000ffu) sv.x = NEG_INF;
            if (mw & 0x0000ff00u) sv.y = NEG_INF;
            if (mw & 0x00ff0000u) sv.z = NEG_INF;
            if (mw & 0xff000000u) sv.w = NEG_INF;
            *(v4f*)&srow[c] = sv;
            mx = fmaxf(mx, fmaxf(fmaxf(sv.x, sv.y), fmaxf(sv.z, sv.w)));
        }
#pragma unroll
        for (int off = 8; off; off >>= 1) mx = fmaxf(mx, __shfl_xor(mx, off, 16));

        float sum = 0.0f;
        for (int j = 0; j < S_LEN / 64; ++j) {
            const int c = 4 * l + 64 * j;
            v4f sv = *(v4f*)&srow[c];
            sv.x = __expf(sv.x - mx);
            sv.y = __expf(sv.y - mx);
            sv.z = __expf(sv.z - mx);
            sv.w = __expf(sv.w - mx);
            *(v4f*)&srow[c] = sv;
            sum += (sv.x + sv.y) + (sv.z + sv.w);
        }
#pragma unroll
        for (int off = 8; off; off >>= 1) sum += __shfl_xor(sum, off, 16);

        const float inv = 1.0f / sum;
        float* arow = attn + ((size_t)b * S_LEN + q0 + r) * S_LEN;
        for (int j = 0; j < S_LEN / 64; ++j) {
            const int c = 4 * l + 64 * j;
            v4f sv = *(v4f*)&srow[c];
            sv.x *= inv; sv.y *= inv; sv.z *= inv; sv.w *= inv;
            *(v4f*)&srow[c] = sv;              // normalized P stays in LDS
            *(v4f*)&arow[c] = sv;              // stream attn to HBM once
        }
    }
    __syncthreads();

    // =========== Phase 3: out = P @ V (P from LDS, V from L2) ===========
    // Each wave owns a K-slice of 256; 4 independent accumulator chains
    // cover D=64 (4 N-tiles of 16).
    {
        v8f acc[4] = {{}, {}, {}, {}};
        const int kbase = wave * 256;
        for (int kk = 0; kk < 64; ++kk) {
            const int k0 = kbase + 4 * kk;
            v2f ap = *(const v2f*)&s_scores[mrow * S_LEN + k0 + c0];
#pragma unroll
            for (int nt = 0; nt < 4; ++nt) {
                const int n = nt * 16 + mrow;
                v2f bv;
                bv.x = vb[(size_t)(k0 + c0) * D_DIM + n];
                bv.y = vb[(size_t)(k0 + c0 + 1) * D_DIM + n];
                acc[nt] = __builtin_amdgcn_wmma_f32_16x16x4_f32(
                    false, ap, false, bv, (short)0, acc[nt], false, false);
            }
        }
        // merge the 8 per-wave partials via LDS float atomics (ds_add_f32)
#pragma unroll
        for (int nt = 0; nt < 4; ++nt)
#pragma unroll
            for (int i = 0; i < 8; ++i)
                atomicAdd(&s_out[(i + mofs) * D_DIM + nt * 16 + mrow], acc[nt][i]);
    }
    __syncthreads();

    // ---- write the 16x64 output tile ----
    float* ob = out + ((size_t)b * S_LEN + q0) * D_DIM;
    for (int i = tid; i < 16 * D_DIM; i += 256)
        ob[i] = s_out[i];
}

extern "C" void kernel_launch(void* const* d_in, const int* in_sizes, int n_in,
                              void* d_out, int out_size, void* d_ws, size_t ws_size,
                              hipStream_t stream) {
    const float*         q    = (const float*)d_in[0];
    const float*         k    = (const float*)d_in[1];
    const float*         v    = (const float*)d_in[2];
    const unsigned char* mask = (const unsigned char*)d_in[3];

    const int B = in_sizes[0] / (S_LEN * D_DIM);   // 16

    float* out  = (float*)d_out;                            // [B,S,D] first
    float* attn = out + (size_t)B * S_LEN * D_DIM;          // then [B,S,S]

    const size_t smem = (size_t)(16 * S_LEN + 2 * 16 * D_DIM) * sizeof(float); // 136KB

    // allow >64KB dynamic LDS (CDNA5 WGP has 320KB)
    hipFuncSetAttribute((const void*)sdpa_f32_wmma_kernel,
                        hipFuncAttributeMaxDynamicSharedMemorySize, (int)smem);

    dim3 grid(B * (S_LEN / 16));   // 2048 workgroups
    sdpa_f32_wmma_kernel<<<grid, 256, smem, stream>>>(q, k, v, mask, out, attn);
}